// PDSSMLayer_52355651338671
// MI455X (gfx1250) — compile-verified
//
#include <hip/hip_runtime.h>
#include <math.h>

// Problem constants (reference: N=64, H=128, K=6, L=8192)
#define L_SEQ   8192
#define H_DIM   128
#define N_ST    64
#define K_SEL   6
#define NCAT    288        // 64 mag + 64 pha + 64 Bre + 64 Bim + 6 sel + 26 pad
#define G_CHUNKS 128
#define T_CHUNK  64        // L_SEQ / G_CHUNKS
#define TWO_PI  6.283185307179586476925f

typedef float v2f __attribute__((ext_vector_type(2)));
typedef float v8f __attribute__((ext_vector_type(8)));

// ---------------------------------------------------------------------------
// Prep: pack weights.
//   Wcat (NCAT x 128): rows 0-63 W1_mag, 64-127 W1_pha, 128-191 B_re,
//                      192-255 B_im, 256-261 P_selector, rest zero.
//   By   (128 x 128) row-major K x N: By[k][h] = C_re[h][k] (k<64)
//                                              = -C_im[h][k-64] (k>=64)
// ---------------------------------------------------------------------------
__global__ void prep_kernel(const float* __restrict__ W1m, const float* __restrict__ W1p,
                            const float* __restrict__ Bre, const float* __restrict__ Bim,
                            const float* __restrict__ Psel,
                            const float* __restrict__ Cre, const float* __restrict__ Cim,
                            float* __restrict__ Wcat, float* __restrict__ By) {
  int gid = blockIdx.x * blockDim.x + threadIdx.x;
  const int wtot = NCAT * H_DIM;
  if (gid < wtot) {
    int r = gid / H_DIM, c = gid % H_DIM;
    float v = 0.f;
    if      (r < 64)  v = W1m[r * H_DIM + c];
    else if (r < 128) v = W1p[(r - 64) * H_DIM + c];
    else if (r < 192) v = Bre[(r - 128) * H_DIM + c];
    else if (r < 256) v = Bim[(r - 192) * H_DIM + c];
    else if (r < 262) v = Psel[(r - 256) * H_DIM + c];
    Wcat[gid] = v;
  } else {
    int j = gid - wtot;
    if (j < H_DIM * H_DIM) {
      int k = j / H_DIM, h = j % H_DIM;
      By[j] = (k < 64) ? Cre[h * N_ST + k] : -Cim[h * N_ST + (k - 64)];
    }
  }
}

// ---------------------------------------------------------------------------
// GEMM1: Z[l, c] = sum_k X[l,k] * W[c,k]   (both operands channel-major, K=128)
// One 16x16 tile per wave, fp32 WMMA (v_wmma_f32_16x16x4_f32), K stepped by 4.
// A 16x4 frag: lane m = lane&15; VGPR0 = K {0 | 2}, VGPR1 = K {1 | 3}.
// ---------------------------------------------------------------------------
__global__ void gemm_xwT(const float* __restrict__ X, const float* __restrict__ W,
                         float* __restrict__ Z, int Kdim, int Ncols) {
  int wave  = threadIdx.x >> 5;
  int lane  = threadIdx.x & 31;
  int ntile = blockIdx.y * (blockDim.x >> 5) + wave;
  if (ntile * 16 >= Ncols) return;               // wave-uniform: EXEC stays all-1s
  int row0 = blockIdx.x * 16;
  int col0 = ntile * 16;
  int mn = lane & 15;
  int kb = (lane >> 4) << 1;                     // 0 or 2
  const float* xa = X + (size_t)(row0 + mn) * Kdim + kb;
  const float* wb = W + (size_t)(col0 + mn) * Kdim + kb;
  v8f acc = {};
  for (int k = 0; k < Kdim; k += 4) {
    v2f a = { xa[k], xa[k + 1] };
    v2f b = { wb[k], wb[k + 1] };
    acc = __builtin_amdgcn_wmma_f32_16x16x4_f32(false, a, false, b,
                                                (short)0, acc, false, false);
  }
  int mb = (lane >> 4) << 3;                     // 0 or 8
  float* zp = Z + (size_t)(row0 + mb) * Ncols + col0 + mn;
#pragma unroll
  for (int r = 0; r < 8; ++r) zp[(size_t)r * Ncols] = acc[r];
}

// ---------------------------------------------------------------------------
// Activations: D_lambda = sigmoid(zm+b) * exp(i * 2pi * sigmoid(zp+b)); Bu pass.
// ---------------------------------------------------------------------------
__global__ void act_kernel(const float* __restrict__ Z,
                           const float* __restrict__ b1m, const float* __restrict__ b1p,
                           float* __restrict__ Dre, float* __restrict__ Dim,
                           float* __restrict__ Bure, float* __restrict__ Buim) {
  int gid = blockIdx.x * blockDim.x + threadIdx.x;     // l*64 + n
  int l = gid >> 6, n = gid & 63;
  const float* zr = Z + (size_t)l * NCAT;
  float zm  = zr[n]      + b1m[n];
  float zp  = zr[64 + n] + b1p[n];
  float mag = 1.f / (1.f + __expf(-zm));
  float ph  = (1.f / (1.f + __expf(-zp))) * TWO_PI;
  float s, c;
  __sincosf(ph, &s, &c);
  Dre[gid]  = mag * c;
  Dim[gid]  = mag * s;
  Bure[gid] = zr[128 + n];
  Buim[gid] = zr[192 + n];
}

// Unnormalized softmax weights over K=6 (argmax-invariant; max-subtracted).
__global__ void sel_kernel(const float* __restrict__ Z, float* __restrict__ selw) {
  int l = blockIdx.x * blockDim.x + threadIdx.x;
  if (l >= L_SEQ) return;
  const float* zr = Z + (size_t)l * NCAT + 256;
  float m = zr[0];
#pragma unroll
  for (int k = 1; k < K_SEL; ++k) m = fmaxf(m, zr[k]);
#pragma unroll
  for (int k = 0; k < K_SEL; ++k) selw[l * K_SEL + k] = __expf(zr[k] - m);
}

// ---------------------------------------------------------------------------
// P_idx[l,n] = argmax_m sum_k w[l,k] * P_dict[k,m,n]   (first max wins, as argmax)
// 256 threads = 4 timesteps x 64 columns; P_dict (96KB) stays L2-resident.
// ---------------------------------------------------------------------------
__global__ void pidx_kernel(const float* __restrict__ Pd,
                            const float* __restrict__ selw, int* __restrict__ Pidx) {
  int tt = threadIdx.x >> 6;
  int n  = threadIdx.x & 63;
  int t  = blockIdx.x * 4 + tt;
  float w[K_SEL];
#pragma unroll
  for (int k = 0; k < K_SEL; ++k) w[k] = selw[t * K_SEL + k];
  float best = -INFINITY;
  int bm = 0;
  for (int m = 0; m < N_ST; ++m) {
    float acc = 0.f;
#pragma unroll
    for (int k = 0; k < K_SEL; ++k)
      acc += w[k] * __ldg(&Pd[((size_t)k * N_ST + m) * N_ST + n]);
    if (acc > best) { best = acc; bm = m; }
  }
  Pidx[(size_t)t * N_ST + n] = bm;
}

// ---------------------------------------------------------------------------
// Phase 1 of scan: per-chunk operator composition q_c = binop(q_c, q_t).
//   P' [n] = Pt[Pc[n]];  D'[n] = Dt[Pc[n]] * Dc[n]
//   b' [m] = Bu_t[m] + sum_{n: Pt[n]==m} Dt[n] * bc[n]   (LDS ds_add_f32 scatter)
// One chunk per 64-thread block, 64 sequential steps.
// ---------------------------------------------------------------------------
__global__ void chunk_reduce(const int* __restrict__ Pidx,
                             const float* __restrict__ Dre, const float* __restrict__ Dim,
                             const float* __restrict__ Bure, const float* __restrict__ Buim,
                             int* __restrict__ CP, float* __restrict__ CDre,
                             float* __restrict__ CDim, float* __restrict__ CBre,
                             float* __restrict__ CBim) {
  __shared__ int   sPt[N_ST];
  __shared__ float sDtre[N_ST], sDtim[N_ST], sTre[N_ST], sTim[N_ST];
  int n = threadIdx.x, g = blockIdx.x;
  int   rP = n;
  float rDre = 1.f, rDim = 0.f, rbre = 0.f, rbim = 0.f;
  int base = g * T_CHUNK;
  for (int s = 0; s < T_CHUNK; ++s) {
    size_t o = (size_t)(base + s) * N_ST + n;
    int   pt = Pidx[o];
    float dr = Dre[o], di = Dim[o];
    sPt[n] = pt; sDtre[n] = dr; sDtim[n] = di;
    sTre[n] = Bure[o]; sTim[n] = Buim[o];
    __syncthreads();
    int   newP = sPt[rP];
    float gr = sDtre[rP], gi = sDtim[rP];
    float nDre = gr * rDre - gi * rDim;
    float nDim = gr * rDim + gi * rDre;
    atomicAdd(&sTre[pt], dr * rbre - di * rbim);
    atomicAdd(&sTim[pt], dr * rbim + di * rbre);
    __syncthreads();
    rP = newP; rDre = nDre; rDim = nDim;
    rbre = sTre[n]; rbim = sTim[n];
    __syncthreads();
  }
  size_t o = (size_t)g * N_ST + n;
  CP[o] = rP; CDre[o] = rDre; CDim[o] = rDim; CBre[o] = rbre; CBim[o] = rbim;
}

// Phase 2: sequential scan over 128 chunk operators; emit exclusive carries.
__global__ void chunk_scan(const int* __restrict__ CP,
                           const float* __restrict__ CDre, const float* __restrict__ CDim,
                           const float* __restrict__ CBre, const float* __restrict__ CBim,
                           float* __restrict__ CarryRe, float* __restrict__ CarryIm) {
  __shared__ float sTre[N_ST], sTim[N_ST];
  int n = threadIdx.x;
  float hre = 0.f, him = 0.f;
  for (int g = 0; g < G_CHUNKS; ++g) {
    size_t o = (size_t)g * N_ST + n;
    CarryRe[o] = hre; CarryIm[o] = him;
    int   p  = CP[o];
    float dr = CDre[o], di = CDim[o];
    sTre[n] = CBre[o]; sTim[n] = CBim[o];
    __syncthreads();
    atomicAdd(&sTre[p], dr * hre - di * him);
    atomicAdd(&sTim[p], dr * him + di * hre);
    __syncthreads();
    hre = sTre[n]; him = sTim[n];
    __syncthreads();
  }
}

// Phase 3: replay each chunk from its carry; write Hcat[l] = [h_re(64) | h_im(64)].
__global__ void chunk_apply(const int* __restrict__ Pidx,
                            const float* __restrict__ Dre, const float* __restrict__ Dim,
                            const float* __restrict__ Bure, const float* __restrict__ Buim,
                            const float* __restrict__ CarryRe, const float* __restrict__ CarryIm,
                            float* __restrict__ Hcat) {
  __shared__ float sTre[N_ST], sTim[N_ST];
  int n = threadIdx.x, g = blockIdx.x;
  float hre = CarryRe[(size_t)g * N_ST + n];
  float him = CarryIm[(size_t)g * N_ST + n];
  int base = g * T_CHUNK;
  for (int s = 0; s < T_CHUNK; ++s) {
    size_t o = (size_t)(base + s) * N_ST + n;
    int   pt = Pidx[o];
    float dr = Dre[o], di = Dim[o];
    sTre[n] = Bure[o]; sTim[n] = Buim[o];
    __syncthreads();
    atomicAdd(&sTre[pt], dr * hre - di * him);
    atomicAdd(&sTim[pt], dr * him + di * hre);
    __syncthreads();
    hre = sTre[n]; him = sTim[n];
    size_t ho = (size_t)(base + s) * H_DIM;
    Hcat[ho + n] = hre;
    Hcat[ho + 64 + n] = him;
    __syncthreads();
  }
}

// ---------------------------------------------------------------------------
// GEMM2: Y = Hcat(Lx128) @ By(128x128)  + D * x   (fused skip epilogue)
// ---------------------------------------------------------------------------
__global__ void gemm_out(const float* __restrict__ Hcat, const float* __restrict__ By,
                         const float* __restrict__ Dvec, const float* __restrict__ X,
                         float* __restrict__ Y) {
  int wave  = threadIdx.x >> 5;
  int lane  = threadIdx.x & 31;
  int ntile = blockIdx.y * (blockDim.x >> 5) + wave;
  if (ntile * 16 >= H_DIM) return;
  int row0 = blockIdx.x * 16, col0 = ntile * 16;
  int mn = lane & 15;
  int kb = (lane >> 4) << 1;
  const float* ap = Hcat + (size_t)(row0 + mn) * H_DIM + kb;
  const float* bp = By + (size_t)kb * H_DIM + col0 + mn;
  v8f acc = {};
  for (int k = 0; k < H_DIM; k += 4) {
    v2f a = { ap[k], ap[k + 1] };
    v2f b = { bp[(size_t)k * H_DIM], bp[(size_t)(k + 1) * H_DIM] };
    acc = __builtin_amdgcn_wmma_f32_16x16x4_f32(false, a, false, b,
                                                (short)0, acc, false, false);
  }
  int mb = (lane >> 4) << 3;
  float dcol = Dvec[col0 + mn];
#pragma unroll
  for (int r = 0; r < 8; ++r) {
    size_t row = (size_t)(row0 + mb + r);
    Y[row * H_DIM + col0 + mn] = acc[r] + dcol * X[row * H_DIM + col0 + mn];
  }
}

// ---------------------------------------------------------------------------
extern "C" void kernel_launch(void* const* d_in, const int* in_sizes, int n_in,
                              void* d_out, int out_size, void* d_ws, size_t ws_size,
                              hipStream_t stream) {
  (void)in_sizes; (void)n_in; (void)out_size; (void)ws_size;
  const float* x    = (const float*)d_in[0];
  const float* B_re = (const float*)d_in[1];
  const float* B_im = (const float*)d_in[2];
  const float* C_re = (const float*)d_in[3];
  const float* C_im = (const float*)d_in[4];
  const float* Dsk  = (const float*)d_in[5];
  const float* Pd   = (const float*)d_in[6];
  const float* Psel = (const float*)d_in[7];
  const float* W1m  = (const float*)d_in[8];
  const float* b1m  = (const float*)d_in[9];
  const float* W1p  = (const float*)d_in[10];
  const float* b1p  = (const float*)d_in[11];
  float* Y = (float*)d_out;

  // Workspace carving (256B aligned slices)
  char* ws = (char*)d_ws;
  size_t off = 0;
  auto alloc = [&](size_t bytes) -> char* {
    char* p = ws + off;
    off += (bytes + 255) & ~(size_t)255;
    return p;
  };
  float* Wcat   = (float*)alloc((size_t)NCAT * H_DIM * 4);
  float* By     = (float*)alloc((size_t)H_DIM * H_DIM * 4);
  float* Z      = (float*)alloc((size_t)L_SEQ * NCAT * 4);
  float* Dre    = (float*)alloc((size_t)L_SEQ * N_ST * 4);
  float* Dim    = (float*)alloc((size_t)L_SEQ * N_ST * 4);
  float* Bure   = (float*)alloc((size_t)L_SEQ * N_ST * 4);
  float* Buim   = (float*)alloc((size_t)L_SEQ * N_ST * 4);
  float* selw   = (float*)alloc((size_t)L_SEQ * K_SEL * 4);
  int*   Pidx   = (int*)  alloc((size_t)L_SEQ * N_ST * 4);
  int*   CP     = (int*)  alloc((size_t)G_CHUNKS * N_ST * 4);
  float* CDre   = (float*)alloc((size_t)G_CHUNKS * N_ST * 4);
  float* CDim   = (float*)alloc((size_t)G_CHUNKS * N_ST * 4);
  float* CBre   = (float*)alloc((size_t)G_CHUNKS * N_ST * 4);
  float* CBim   = (float*)alloc((size_t)G_CHUNKS * N_ST * 4);
  float* CarRe  = (float*)alloc((size_t)G_CHUNKS * N_ST * 4);
  float* CarIm  = (float*)alloc((size_t)G_CHUNKS * N_ST * 4);
  float* Hcat   = (float*)alloc((size_t)L_SEQ * H_DIM * 4);

  // 0. pack weights
  {
    int tot = NCAT * H_DIM + H_DIM * H_DIM;
    prep_kernel<<<(tot + 255) / 256, 256, 0, stream>>>(W1m, W1p, B_re, B_im, Psel,
                                                       C_re, C_im, Wcat, By);
  }
  // 1. fused input projection GEMM (WMMA fp32): Z = x @ Wcat^T
  gemm_xwT<<<dim3(L_SEQ / 16, 5), 128, 0, stream>>>(x, Wcat, Z, H_DIM, NCAT);
  // 2. activations / selector weights
  act_kernel<<<(L_SEQ * N_ST) / 256, 256, 0, stream>>>(Z, b1m, b1p, Dre, Dim, Bure, Buim);
  sel_kernel<<<L_SEQ / 256, 256, 0, stream>>>(Z, selw);
  // 3. hard permutation indices
  pidx_kernel<<<L_SEQ / 4, 256, 0, stream>>>(Pd, selw, Pidx);
  // 4-6. three-phase chunked parallel scan
  chunk_reduce<<<G_CHUNKS, N_ST, 0, stream>>>(Pidx, Dre, Dim, Bure, Buim,
                                              CP, CDre, CDim, CBre, CBim);
  chunk_scan<<<1, N_ST, 0, stream>>>(CP, CDre, CDim, CBre, CBim, CarRe, CarIm);
  chunk_apply<<<G_CHUNKS, N_ST, 0, stream>>>(Pidx, Dre, Dim, Bure, Buim,
                                             CarRe, CarIm, Hcat);
  // 7. output GEMM (WMMA fp32) + D*x skip
  gemm_out<<<dim3(L_SEQ / 16, 2), 128, 0, stream>>>(Hcat, By, Dsk, x, Y);
}